// YOLOLoss_66374424592688
// MI455X (gfx1250) — compile-verified
//
#include <hip/hip_runtime.h>
#include <math.h>

// ---- constants from the reference ----
#define NCLS 80
#define NANC 3
#define NLVL 3
#define CH   (5 + NCLS)   // 85
#define GBIAS 0.5f
#define EPSF  1e-7f

#if defined(__gfx1250__) && __has_builtin(__builtin_amdgcn_global_load_async_to_lds_b32) && \
    __has_builtin(__builtin_amdgcn_s_wait_asynccnt)
#define HAVE_ASYNC_LDS 1
#else
#define HAVE_ASYNC_LDS 0
#endif

typedef __attribute__((ext_vector_type(2))) float v2f;
typedef __attribute__((ext_vector_type(8))) float v8f;

#if HAVE_ASYNC_LDS
typedef __attribute__((address_space(1))) int as1_int;  // global
typedef __attribute__((address_space(3))) int as3_int;  // LDS
#endif

__device__ __forceinline__ float sigmoidf_(float x) { return 1.0f / (1.0f + expf(-x)); }
// bce_logits(x,y) = softplus(x) - x*y ; with y==0 it is softplus(x)
__device__ __forceinline__ float softplus_(float x) {
    return fmaxf(x, 0.0f) + log1pf(expf(-fabsf(x)));
}

// ---------------- workspace layout (floats) ----------------
// acc[0..2]  : per-level lbox sums
// acc[3..5]  : per-level lcls sums
// acc[6..8]  : per-level lobj sums
// ((uint*)acc)[12..14] : per-level mask counts (nv)
// acc[16...] : tobj scatter buffers, level 0 then 1 then 2

__global__ void k_zero(float* ws, long n) {
    long i = blockIdx.x * (long)blockDim.x + threadIdx.x;
    long stride = (long)gridDim.x * blockDim.x;
    for (; i < n; i += stride) ws[i] = 0.0f;
}

__global__ void k_targets(const float* __restrict__ p0,
                          const float* __restrict__ p1,
                          const float* __restrict__ p2,
                          const float* __restrict__ tgt,
                          const float* __restrict__ anc,
                          float* acc,
                          float* tobj0, float* tobj1, float* tobj2,
                          int bs, int nt)
{
    int idx = blockIdx.x * blockDim.x + threadIdx.x;
    int total = NLVL * 5 * NANC * nt;
    if (idx >= total) return;

    int t   = idx % nt;
    int a   = (idx / nt) % NANC;
    int o   = (idx / (nt * NANC)) % 5;
    int lvl = idx / (nt * NANC * 5);

    int Wg = (lvl == 0) ? 80 : ((lvl == 1) ? 40 : 20);
    int Hg = Wg;
    const float* pi = (lvl == 0) ? p0 : ((lvl == 1) ? p1 : p2);
    float* tobj     = (lvl == 0) ? tobj0 : ((lvl == 1) ? tobj1 : tobj2);

    float img = tgt[t * 6 + 0];
    float cls = tgt[t * 6 + 1];
    float gx  = tgt[t * 6 + 2] * (float)Wg;
    float gy  = tgt[t * 6 + 3] * (float)Hg;
    float gw  = tgt[t * 6 + 4] * (float)Wg;
    float gh  = tgt[t * 6 + 5] * (float)Hg;
    float aw  = anc[(lvl * NANC + a) * 2 + 0];
    float ah  = anc[(lvl * NANC + a) * 2 + 1];

    // anchor-ratio filter m0
    float rw = gw / aw, rh = gh / ah;
    float m0v = fmaxf(fmaxf(rw, 1.0f / rw), fmaxf(rh, 1.0f / rh));
    bool m0 = (m0v < 4.0f);

    // offset-row condition + offset
    bool jm = true;
    float offx = 0.0f, offy = 0.0f;
    if (o == 1)      { jm = (fmodf(gx, 1.0f) < GBIAS) && (gx > 1.0f); offx =  GBIAS; }
    else if (o == 2) { jm = (fmodf(gy, 1.0f) < GBIAS) && (gy > 1.0f); offy =  GBIAS; }
    else if (o == 3) { float gxi = (float)Wg - gx;
                       jm = (fmodf(gxi, 1.0f) < GBIAS) && (gxi > 1.0f); offx = -GBIAS; }
    else if (o == 4) { float gyi = (float)Hg - gy;
                       jm = (fmodf(gyi, 1.0f) < GBIAS) && (gyi > 1.0f); offy = -GBIAS; }

    if (!(jm && m0)) return;

    atomicAdd((unsigned int*)acc + 12 + lvl, 1u);  // nv count

    int gi = (int)truncf(gx - offx);
    int gj = (int)truncf(gy - offy);
    gi = min(max(gi, 0), Wg - 1);
    gj = min(max(gj, 0), Hg - 1);

    float tx = gx - (float)gi;
    float ty = gy - (float)gj;
    float tw = gw, th = gh;

    int b = (int)img;
    int c = (int)cls;

    long base = ((((long)b * NANC + a) * Hg + gj) * Wg + gi) * CH;
    const float* ps = pi + base;

    float px = sigmoidf_(ps[0]) * 2.0f - 0.5f;
    float py = sigmoidf_(ps[1]) * 2.0f - 0.5f;
    float pw = sigmoidf_(ps[2]) * 2.0f; pw = pw * pw * aw;
    float ph = sigmoidf_(ps[3]) * 2.0f; ph = ph * ph * ah;

    // ----- CIoU(pbox, tbox), matching reference EPS placement -----
    float b1x1 = px - pw * 0.5f, b1x2 = px + pw * 0.5f;
    float b1y1 = py - ph * 0.5f, b1y2 = py + ph * 0.5f;
    float b2x1 = tx - tw * 0.5f, b2x2 = tx + tw * 0.5f;
    float b2y1 = ty - th * 0.5f, b2y2 = ty + th * 0.5f;

    float inter = fmaxf(fminf(b1x2, b2x2) - fmaxf(b1x1, b2x1), 0.0f) *
                  fmaxf(fminf(b1y2, b2y2) - fmaxf(b1y1, b2y1), 0.0f);
    float w1 = b1x2 - b1x1, h1 = b1y2 - b1y1 + EPSF;
    float w2 = b2x2 - b2x1, h2 = b2y2 - b2y1 + EPSF;
    float uni = w1 * h1 + w2 * h2 - inter + EPSF;
    float iou = inter / uni;

    float cw  = fmaxf(b1x2, b2x2) - fminf(b1x1, b2x1);
    float chh = fmaxf(b1y2, b2y2) - fminf(b1y1, b2y1);
    float c2  = cw * cw + chh * chh + EPSF;
    float dx  = b2x1 + b2x2 - b1x1 - b1x2;
    float dy  = b2y1 + b2y2 - b1y1 - b1y2;
    float rho2 = (dx * dx + dy * dy) * 0.25f;
    float dv = atanf(w2 / h2) - atanf(w1 / h1);
    const float k4pi2 = 4.0f / (3.14159265358979323846f * 3.14159265358979323846f);
    float v = k4pi2 * dv * dv;
    float alpha = v / (v - iou + (1.0f + EPSF));
    float ciou = iou - (rho2 / c2 + v * alpha);

    atomicAdd(&acc[lvl], 1.0f - ciou);

    // obj target scatter (GR==1 -> clip(iou,0)); plain store == .set semantics
    tobj[(((long)b * NANC + a) * Hg + gj) * Wg + gi] = fmaxf(ciou, 0.0f);

    // class BCE vs one-hot: sum_k softplus(ps[5+k]) - ps[5+c]
    float s = 0.0f;
    #pragma unroll 4
    for (int k = 0; k < NCLS; ++k) s += softplus_(ps[5 + k]);
    s -= ps[5 + c];
    atomicAdd(&acc[3 + lvl], s);
}

// lobj partial: sum over cells of bce(p[...,4], tobj).
// Stride-85 objectness gather is staged through LDS with async global->LDS
// loads (ASYNCcnt, double-buffered, exact per-wave issue counting), then the
// block reduction compresses 256 partials via LDS to one float2 per lane of
// wave 0 and a single V_WMMA_F32_16X16X4_F32 (all-ones B => row sums).
__global__ void __launch_bounds__(256) k_obj(const float* __restrict__ pi,
                                             const float* __restrict__ tobj,
                                             float* acc_slot, long ncells)
{
    __shared__ float sdata[256];
    long i0 = blockIdx.x * (long)blockDim.x + threadIdx.x;
    long stride = (long)gridDim.x * blockDim.x;
    float local = 0.0f;

#if HAVE_ASYNC_LDS
    __shared__ float stage[2][256];
    // Prologue: issue batch 0 (clamped address so every wave issues exactly once).
    {
        long c0 = (i0 < ncells) ? i0 : (ncells - 1);
        __builtin_amdgcn_global_load_async_to_lds_b32(
            (as1_int*)(pi + c0 * CH + 4),
            (as3_int*)&stage[0][threadIdx.x],
            0, 0);
    }
    int buf = 0;
    for (long i = i0; i < ncells; i += stride) {
        long inext = i + stride;
        long cn = (inext < ncells) ? inext : (ncells - 1);
        // Issue batch k+1 into the other buffer (always issues: clamped addr).
        __builtin_amdgcn_global_load_async_to_lds_b32(
            (as1_int*)(pi + cn * CH + 4),
            (as3_int*)&stage[buf ^ 1][threadIdx.x],
            0, 0);
        // Wait until only the just-issued batch is outstanding -> batch k ready.
        __builtin_amdgcn_s_wait_asynccnt(1);
        float x = stage[buf][threadIdx.x];
        float y = tobj[i];
        local += fmaxf(x, 0.0f) - x * y + log1pf(expf(-fabsf(x)));
        buf ^= 1;
    }
    // Trailing clamped issue is drained by S_ENDPGM's implicit wait-idle.
#else
    for (long i = i0; i < ncells; i += stride) {
        float x = pi[i * CH + 4];
        float y = tobj[i];
        local += fmaxf(x, 0.0f) - x * y + log1pf(expf(-fabsf(x)));
    }
#endif

    sdata[threadIdx.x] = local;
    __syncthreads();

    if (threadIdx.x < 32) {   // exactly wave 0, EXEC all-ones inside
        int l = threadIdx.x;
        v2f A;
        A.x = sdata[l * 8 + 0] + sdata[l * 8 + 1] + sdata[l * 8 + 2] + sdata[l * 8 + 3];
        A.y = sdata[l * 8 + 4] + sdata[l * 8 + 5] + sdata[l * 8 + 6] + sdata[l * 8 + 7];
        v2f B; B.x = 1.0f; B.y = 1.0f;          // all-ones B => row sums
        v8f C = {};
        C = __builtin_amdgcn_wmma_f32_16x16x4_f32(
                false, A, false, B, (short)0, C, false, false);
        // lanes 0-15 hold rows 0-7 in C[0..7]; lanes 16-31 hold rows 8-15
        float s = C[0] + C[1] + C[2] + C[3] + C[4] + C[5] + C[6] + C[7];
        float tot = __shfl(s, 0, 32) + __shfl(s, 16, 32);
        if (l == 0) atomicAdd(acc_slot, tot);
    }
}

__global__ void k_final(const float* acc, float* out, int bs) {
    if (threadIdx.x != 0 || blockIdx.x != 0) return;
    const unsigned int* cnt = (const unsigned int*)acc + 12;
    const int Ws[3] = {80, 40, 20};
    float lbox = 0.0f, lcls = 0.0f, lobj = 0.0f;
    for (int i = 0; i < 3; ++i) {
        float nv = fmaxf((float)cnt[i], 1.0f);
        lbox += acc[i] / nv;
        lcls += acc[3 + i] / (nv * (float)NCLS);
        float cells = (float)bs * (float)NANC * (float)(Ws[i] * Ws[i]);
        lobj += acc[6 + i] / cells;   // per-level mean, OBJ_GAIN == 1
    }
    lbox *= 0.05f;   // BOX_GAIN
    lcls *= 0.5f;    // CLS_GAIN
    // lobj *= 1.0f  // OBJ_GAIN
    float loss = (lbox + lobj + lcls) * (float)bs;
    out[0] = loss; out[1] = lbox; out[2] = lobj; out[3] = lcls;
}

extern "C" void kernel_launch(void* const* d_in, const int* in_sizes, int n_in,
                              void* d_out, int out_size, void* d_ws, size_t ws_size,
                              hipStream_t stream)
{
    const float* p0  = (const float*)d_in[0];
    const float* p1  = (const float*)d_in[1];
    const float* p2  = (const float*)d_in[2];
    const float* tgt = (const float*)d_in[3];
    const float* anc = (const float*)d_in[4];

    int bs = in_sizes[0] / (NANC * 80 * 80 * CH);
    int nt = in_sizes[3] / 6;

    float* ws   = (float*)d_ws;
    float* acc  = ws;              // 16-float header
    long c0 = (long)bs * NANC * 80 * 80;
    long c1 = (long)bs * NANC * 40 * 40;
    long c2 = (long)bs * NANC * 20 * 20;
    float* tobj0 = ws + 16;
    float* tobj1 = tobj0 + c0;
    float* tobj2 = tobj1 + c1;

    long ztotal = 16 + c0 + c1 + c2;
    int zblocks = (int)((ztotal + 255) / 256);
    k_zero<<<zblocks, 256, 0, stream>>>(ws, ztotal);

    int total_t = NLVL * 5 * NANC * nt;
    k_targets<<<(total_t + 255) / 256, 256, 0, stream>>>(
        p0, p1, p2, tgt, anc, acc, tobj0, tobj1, tobj2, bs, nt);

    long b0 = (c0 + 255) / 256; if (b0 > 1024) b0 = 1024;
    long b1 = (c1 + 255) / 256; if (b1 > 1024) b1 = 1024;
    long b2 = (c2 + 255) / 256; if (b2 > 1024) b2 = 1024;
    k_obj<<<(int)b0, 256, 0, stream>>>(p0, tobj0, acc + 6, c0);
    k_obj<<<(int)b1, 256, 0, stream>>>(p1, tobj1, acc + 7, c1);
    k_obj<<<(int)b2, 256, 0, stream>>>(p2, tobj2, acc + 8, c2);

    k_final<<<1, 1, 0, stream>>>(acc, (float*)d_out, bs);
}